// ESIM_65017214926875
// MI455X (gfx1250) — compile-verified
//
#include <hip/hip_runtime.h>
#include <hip/hip_bf16.h>

typedef __attribute__((ext_vector_type(2))) float v2f;
typedef __attribute__((ext_vector_type(8))) float v8f;

#define NND 255      // tree nodes
#define NP  256      // padded nodes
#define MP  320      // padded mem dim (300 -> 20*16, divisible by 64)
#define XGW 1280     // 4 gates * 320
#define KH  640      // lh(320) ++ rh(320)
#define GW  1600     // 5 gates * 320
#define EMB 300
#define MU  300
#define NLVL 8
#define KC  32       // GEMM K chunk staged in LDS

__device__ __forceinline__ float sigf(float x) { return 1.0f / (1.0f + __expf(-x)); }

// B operands are stored K-pair swizzled so a lane's (k, k+1) pair is contiguous:
//   B'[((k>>1)*ld + n)*2 + (k&1)]  ->  compute reads one ds_load_b64, no repacking.
__device__ __forceinline__ size_t bswz(int k, int n, int ld) {
  return ((size_t)(k >> 1) * ld + n) * 2 + (k & 1);
}

// -------------------------------------------------- async global->LDS helpers
#if defined(__gfx1250__) && __has_builtin(__builtin_amdgcn_global_load_async_to_lds_b128)
#define ASYNC_LDS 1
#endif

typedef int vi4 __attribute__((vector_size(16)));
typedef __attribute__((address_space(1))) vi4* gvi4p;   // global (device) AS
typedef __attribute__((address_space(3))) vi4* lvi4p;   // LDS AS

__device__ __forceinline__ void async_copy_b128(float* lds, const float* g) {
#ifdef ASYNC_LDS
  __builtin_amdgcn_global_load_async_to_lds_b128(
      (gvi4p)(vi4*)(const_cast<float*>(g)), (lvi4p)(vi4*)lds, 0, 0);
#else
  *(float4*)lds = *(const float4*)g;
#endif
}

__device__ __forceinline__ void async_wait0() {
#ifdef ASYNC_LDS
#if __has_builtin(__builtin_amdgcn_s_wait_asynccnt)
  __builtin_amdgcn_s_wait_asynccnt(0);
#else
  asm volatile("s_wait_asynccnt 0x0" ::: "memory");
#endif
#endif
}

// ---------------------------------------------------------------- utilities
__global__ void zero_kernel(float* __restrict__ p, long n) {
  long i = (long)blockIdx.x * blockDim.x + threadIdx.x;
  long stride = (long)gridDim.x * blockDim.x;
  for (; i < n; i += stride) p[i] = 0.0f;
}

// ------------------------------------------------ fp32 WMMA GEMM, 128x64 block tile
// Block = 128 threads (4 waves). Wave w computes 2 M-frags (rows m0+16w and m0+64+16w)
// x 4 N-frags = 8 WMMA accumulators; each B ds_load_b64 feeds 2 wmma, each A load 4.
// B (pair-swizzled, see bswz) is double-buffered in LDS via GLOBAL_LOAD_ASYNC_TO_LDS_B128
// and drained with s_wait_asynccnt. Requires K % KC == 0, N % 64 == 0, M % 128 == 0.
// Fragment layouts per CDNA5 ISA 7.12.2 (32-bit A 16x4, B 4x16, C 16x16).
__global__ __launch_bounds__(128) void gemm_f32_wmma(
    const float* __restrict__ A, const float* __restrict__ B, float* __restrict__ C,
    int K, int lda, int ldb, int ldc) {
  __shared__ float Bs[2][KC / 2][132];  // pair-rows of 128 floats, padded to 132 (16B-aligned)
  const int t    = threadIdx.x;
  const int wave = t >> 5;
  const int lane = t & 31;
  const int half = lane >> 4;       // 0: K pair {0,1}; 1: K pair {2,3}
  const int l16  = lane & 15;
  const int m0 = blockIdx.y * 128 + wave * 16;
  const int n0 = blockIdx.x * 64;
  const float* Arow0 = A + (size_t)(m0 + l16) * lda;
  const float* Arow1 = A + (size_t)(m0 + 64 + l16) * lda;

  auto issue_chunk = [&](int buf, int kc0) {
    const int p0 = kc0 >> 1;
#pragma unroll
    for (int i = 0; i < 4; ++i) {
      int e    = t + i * 128;         // 0..511 -> 512 x b128 = 16 pair-rows x 128 floats
      int prow = e >> 5;
      int c4   = (e & 31) << 2;
      async_copy_b128(&Bs[buf][prow][c4],
                      B + (size_t)(p0 + prow) * 2 * ldb + 2 * n0 + c4);
    }
  };

  v8f acc[2][4] = {};
  issue_chunk(0, 0);
  const int nchunks = K / KC;
  for (int c = 0; c < nchunks; ++c) {
    async_wait0();
    __syncthreads();                       // chunk c resident in LDS for all waves
    if (c + 1 < nchunks) issue_chunk((c + 1) & 1, (c + 1) * KC);
    const int cur = c & 1;
    const int kbase = c * KC;
#pragma unroll
    for (int k = 0; k < KC; k += 4) {
      const int ka = k + half * 2;
      const int p  = ka >> 1;              // pair-row within chunk
      v2f a0 = *(const v2f*)(Arow0 + kbase + ka);   // A: lane M=l16, K=ka, ka+1
      v2f a1 = *(const v2f*)(Arow1 + kbase + ka);
#pragma unroll
      for (int nb = 0; nb < 4; ++nb) {
        v2f b = *(const v2f*)&Bs[cur][p][2 * (nb * 16 + l16)];
        acc[0][nb] = __builtin_amdgcn_wmma_f32_16x16x4_f32(false, a0, false, b, (short)0,
                                                           acc[0][nb], false, false);
        acc[1][nb] = __builtin_amdgcn_wmma_f32_16x16x4_f32(false, a1, false, b, (short)0,
                                                           acc[1][nb], false, false);
      }
    }
  }
#pragma unroll
  for (int mi = 0; mi < 2; ++mi) {
#pragma unroll
    for (int nb = 0; nb < 4; ++nb) {
      float* Crow = C + (size_t)(m0 + mi * 64 + half * 8) * ldc + n0 + nb * 16 + l16;
#pragma unroll
      for (int v = 0; v < 8; ++v) Crow[(size_t)v * ldc] = acc[mi][nb][v];
    }
  }
}

// ---------------------------------------------------------------- weight staging
// Bx'[bswz(d, g*MP+m, XGW)] = Wx[g][m][d]    (Wx: [4,300,300])
__global__ void stage_wx_kernel(const float* __restrict__ Wx, float* __restrict__ Bx) {
  int idx = blockIdx.x * blockDim.x + threadIdx.x;
  if (idx >= 4 * MU * EMB) return;
  int d = idx % EMB;
  int m = (idx / EMB) % MU;
  int g = idx / (MU * EMB);
  Bx[bswz(d, g * MP + m, XGW)] = Wx[idx];
}

// Bh'[bswz(d, col, GW)] = Wl[g][m][d];  Bh'[bswz(320+d, col, GW)] = Wr[g][m][d]
__global__ void stage_wh_kernel(const float* __restrict__ Wl, const float* __restrict__ Wr,
                                float* __restrict__ Bh) {
  int idx = blockIdx.x * blockDim.x + threadIdx.x;
  if (idx >= 5 * MU * EMB) return;
  int d = idx % EMB;
  int m = (idx / EMB) % MU;
  int g = idx / (MU * EMB);
  Bh[bswz(d, g * MP + m, GW)]      = Wl[idx];
  Bh[bswz(MP + d, g * MP + m, GW)] = Wr[idx];
}

// BWc'[bswz(k, m, MP)] = Wc[m][k]   (Wc: [300,1200])
__global__ void stage_wc_kernel(const float* __restrict__ Wc, float* __restrict__ BWc) {
  int idx = blockIdx.x * blockDim.x + threadIdx.x;
  if (idx >= MU * 4 * MU) return;
  int k = idx % (4 * MU);
  int m = idx / (4 * MU);
  BWc[bswz(k, m, MP)] = Wc[idx];
}

// ---------------------------------------------------------------- tree level schedule
__global__ void levels_kernel(const int* __restrict__ left, const int* __restrict__ right,
                              int n, int* __restrict__ counts, int* __restrict__ nodes) {
  __shared__ int depth[NP];
  if (threadIdx.x != 0) return;
  for (int l = 0; l < 16; ++l) counts[l] = 0;
  for (int i = 0; i < n; ++i) {
    int li = left[i];
    int d = 0;
    if (li >= 0) {
      int dl = depth[li];
      int dr = depth[right[i]];
      d = (dl > dr ? dl : dr) + 1;
    }
    if (d > 15) d = 15;
    depth[i] = d;
    int c = counts[d];
    if (c < 128) { nodes[d * 128 + c] = i; counts[d] = c + 1; }
  }
}

// ---------------------------------------------------------------- embedding gather
__global__ void embed_kernel(const int* __restrict__ sent, const int* __restrict__ left,
                             const float* __restrict__ emb, int pad_tok,
                             float* __restrict__ X) {
  int nn = blockIdx.x;       // 0..255
  int t  = threadIdx.x;      // 0..319
  float v = 0.0f;
  if (nn < NND && t < EMB) {
    int tok = (left[nn] < 0) ? sent[nn] : pad_tok;
    v = emb[(size_t)tok * EMB + t];
  }
  X[nn * MP + t] = v;
}

// ---------------------------------------------------------------- tree-LSTM pieces
// A[r][0..299]=H[left], A[r][320..619]=H[right] (zeros elsewhere / r>=count)
__global__ void build_a_kernel(const int* __restrict__ counts, const int* __restrict__ nodes,
                               int lvl, const int* __restrict__ left,
                               const int* __restrict__ right, const float* __restrict__ H,
                               float* __restrict__ A) {
  int r = blockIdx.x;        // 0..127
  int t = threadIdx.x;       // 0..639
  float v = 0.0f;
  if (r < counts[lvl]) {
    int nn = nodes[lvl * 128 + r];
    if (t < MU)                    { int l = left[nn];  if (l >= 0) v = H[l * MP + t]; }
    else if (t >= MP && t < MP+MU) { int q = right[nn]; if (q >= 0) v = H[q * MP + (t - MP)]; }
  }
  A[r * KH + t] = v;
}

__global__ void gate_kernel(const int* __restrict__ counts, const int* __restrict__ nodes,
                            int lvl, const int* __restrict__ left,
                            const int* __restrict__ right, const float* __restrict__ xg,
                            const float* __restrict__ bx, const float* __restrict__ G,
                            float* __restrict__ C, float* __restrict__ H) {
  int r = blockIdx.x;
  int m = threadIdx.x;
  if (m >= MU) return;
  if (r >= counts[lvl]) return;
  int nn = nodes[lvl * 128 + r];
  const float* xr = xg + (size_t)nn * XGW;
  const float* gr = G + (size_t)r * GW;
  float x0 = xr[0 * MP + m] + bx[0 * MU + m];
  float x1 = xr[1 * MP + m] + bx[1 * MU + m];
  float x2 = xr[2 * MP + m] + bx[2 * MU + m];
  float x3 = xr[3 * MP + m] + bx[3 * MU + m];
  float u  = tanhf(x0 + gr[0 * MP + m]);
  float ig = sigf(x1 + gr[1 * MP + m]);
  float lf = sigf(x2 + gr[2 * MP + m]);   // fx shared for both forget gates
  float rf = sigf(x2 + gr[3 * MP + m]);
  float o  = sigf(x3 + gr[4 * MP + m]);
  int l = left[nn], q = right[nn];
  float lc = (l >= 0) ? C[l * MP + m] : 0.0f;
  float rc = (q >= 0) ? C[q * MP + m] : 0.0f;
  float c = ig * u + lf * lc + rf * rc;
  C[nn * MP + m] = c;
  H[nn * MP + m] = o * tanhf(c);
}

// ---------------------------------------------------------------- attention
// H2T'[bswz(d, nn, NP)] = H[nn][d]   (transposed + pair-swizzled, B of raw GEMM)
__global__ void transpose_kernel(const float* __restrict__ H, float* __restrict__ HT) {
  int i = blockIdx.x * blockDim.x + threadIdx.x;   // over MP*NP
  if (i >= MP * NP) return;
  int d = i / NP, nn = i % NP;
  HT[bswz(d, nn, NP)] = H[nn * MP + d];
}

// HS'[bswz(j, m, MP)] = H[j][m]   (pair-swizzled copy, B of beta/alpha GEMMs)
__global__ void hswz_kernel(const float* __restrict__ H, float* __restrict__ HS) {
  int i = blockIdx.x * blockDim.x + threadIdx.x;   // over NP*MP
  if (i >= NP * MP) return;
  int j = i / MP, m = i % MP;
  HS[bswz(j, m, MP)] = H[i];
}

__global__ void softmax_rows_kernel(const float* __restrict__ raw, float* __restrict__ P) {
  int i = blockIdx.x;       // row
  int t = threadIdx.x;      // 256 threads
  __shared__ float red[256];
  float v = (i < NND && t < NND) ? raw[i * NP + t] : -__builtin_inff();
  red[t] = v; __syncthreads();
  for (int s = 128; s > 0; s >>= 1) { if (t < s) red[t] = fmaxf(red[t], red[t + s]); __syncthreads(); }
  float mx = red[0]; __syncthreads();
  float e = (i < NND && t < NND) ? __expf(v - mx) : 0.0f;
  red[t] = e; __syncthreads();
  for (int s = 128; s > 0; s >>= 1) { if (t < s) red[t] += red[t + s]; __syncthreads(); }
  float sum = red[0];
  P[i * NP + t] = (i < NND && sum > 0.0f) ? e / sum : 0.0f;
}

__global__ void softmax_cols_kernel(const float* __restrict__ raw, float* __restrict__ P) {
  int j = blockIdx.x;       // column of raw == row of raw.T
  int t = threadIdx.x;
  __shared__ float red[256];
  float v = (j < NND && t < NND) ? raw[t * NP + j] : -__builtin_inff();
  red[t] = v; __syncthreads();
  for (int s = 128; s > 0; s >>= 1) { if (t < s) red[t] = fmaxf(red[t], red[t + s]); __syncthreads(); }
  float mx = red[0]; __syncthreads();
  float e = (j < NND && t < NND) ? __expf(v - mx) : 0.0f;
  red[t] = e; __syncthreads();
  for (int s = 128; s > 0; s >>= 1) { if (t < s) red[t] += red[t + s]; __syncthreads(); }
  float sum = red[0];
  P[j * NP + t] = (j < NND && sum > 0.0f) ? e / sum : 0.0f;
}

// ---------------------------------------------------------------- compare stage
// feat = [s, a, s-a, s*a], contiguous 1200 cols + zero pad to 1280
__global__ void feat_kernel(const float* __restrict__ H, const float* __restrict__ A,
                            float* __restrict__ F) {
  int nn = blockIdx.x;
  for (int k = threadIdx.x; k < XGW; k += blockDim.x) {
    float v = 0.0f;
    if (k < 4 * MU) {
      int seg = k / MU, m = k % MU;
      float s = H[nn * MP + m];
      float a = A[nn * MP + m];
      v = (seg == 0) ? s : (seg == 1) ? a : (seg == 2) ? (s - a) : (s * a);
    }
    F[(size_t)nn * XGW + k] = v;
  }
}

__global__ void relubias_kernel(const float* __restrict__ G, const float* __restrict__ bc,
                                float* __restrict__ Hc) {
  int nn = blockIdx.x;
  int m = threadIdx.x;
  float v = 0.0f;
  if (nn < NND && m < MU) v = fmaxf(G[nn * MP + m] + bc[m], 0.0f);
  Hc[nn * MP + m] = v;
}

// ---------------------------------------------------------------- pooling head
__global__ void head_kernel(const float* __restrict__ VH1, const float* __restrict__ VH2,
                            const float* __restrict__ Wa1, const float* __restrict__ ba1,
                            const float* __restrict__ Wa2, const float* __restrict__ ba2,
                            float* __restrict__ out) {
  __shared__ float agg[4 * MU];
  __shared__ float hid[MU];
  int t = threadIdx.x;
  for (int k = t; k < 4 * MU; k += blockDim.x) {
    int seg = k / MU, m = k % MU;
    const float* V = (seg < 2) ? VH1 : VH2;
    bool ismax = (seg & 1);
    float acc = ismax ? -__builtin_inff() : 0.0f;
    for (int nn = 0; nn < NND; ++nn) {
      float x = V[nn * MP + m];
      acc = ismax ? fmaxf(acc, x) : (acc + x);
    }
    agg[k] = ismax ? acc : acc / (float)NND;
  }
  __syncthreads();
  for (int m = t; m < MU; m += blockDim.x) {
    float s = ba1[m];
    for (int k = 0; k < 4 * MU; ++k) s += Wa1[(size_t)m * 4 * MU + k] * agg[k];
    hid[m] = fmaxf(s, 0.0f);
  }
  __syncthreads();
  if (t < 3) {
    float s = ba2[t];
    for (int m = 0; m < MU; ++m) s += Wa2[t * MU + m] * hid[m];
    out[t] = s;
  }
}

// ================================================================ launch
extern "C" void kernel_launch(void* const* d_in, const int* in_sizes, int n_in,
                              void* d_out, int out_size, void* d_ws, size_t ws_size,
                              hipStream_t stream) {
  const int*   sent1 = (const int*)d_in[0];
  const int*   sent2 = (const int*)d_in[1];
  const int*   left1 = (const int*)d_in[2];
  const int*   right1= (const int*)d_in[3];
  const int*   left2 = (const int*)d_in[4];
  const int*   right2= (const int*)d_in[5];
  const float* emb   = (const float*)d_in[6];
  const float* Wx_i  = (const float*)d_in[7];
  const float* bx_i  = (const float*)d_in[8];
  const float* Wl_i  = (const float*)d_in[9];
  const float* Wr_i  = (const float*)d_in[10];
  const float* Wx_c  = (const float*)d_in[11];
  const float* bx_c  = (const float*)d_in[12];
  const float* Wl_c  = (const float*)d_in[13];
  const float* Wr_c  = (const float*)d_in[14];
  const float* Wc    = (const float*)d_in[15];
  const float* bc    = (const float*)d_in[16];
  const float* Wa1   = (const float*)d_in[17];
  const float* ba1   = (const float*)d_in[18];
  const float* Wa2   = (const float*)d_in[19];
  const float* ba2   = (const float*)d_in[20];
  float* out = (float*)d_out;

  const int V = in_sizes[6] / EMB;
  const int pad_tok = V - 1;

  // ---- workspace bump allocator (floats) ----
  float* w = (float*)d_ws;
  size_t off = 0;
  auto alloc = [&](size_t n) { float* p = w + off; off += n; return p; };
  float* BXI = alloc((size_t)MP * XGW);   // x-proj weights (input composer), swizzled
  float* BXC = alloc((size_t)MP * XGW);   // x-proj weights (compare composer), swizzled
  float* BHI = alloc((size_t)KH * GW);    // [Wl;Wr] input composer, swizzled
  float* BHC = alloc((size_t)KH * GW);    // [Wl;Wr] compare composer, swizzled
  float* BWC = alloc((size_t)XGW * MP);   // Wc^T, swizzled
  float* X1  = alloc((size_t)NP * MP);
  float* X2  = alloc((size_t)NP * MP);
  float* XG1 = alloc((size_t)NP * XGW);
  float* XG2 = alloc((size_t)NP * XGW);
  float* H1  = alloc((size_t)NP * MP);
  float* C1  = alloc((size_t)NP * MP);
  float* H2  = alloc((size_t)NP * MP);
  float* C2  = alloc((size_t)NP * MP);
  float* ALV = alloc((size_t)128 * KH);
  float* GLV = alloc((size_t)128 * GW);
  float* H2T = alloc((size_t)MP * NP);    // H2 transposed, swizzled
  float* H1S = alloc((size_t)NP * MP);    // H1 pair-swizzled (B of alpha)
  float* H2S = alloc((size_t)NP * MP);    // H2 pair-swizzled (B of beta)
  float* RAW = alloc((size_t)NP * NP);
  float* P1  = alloc((size_t)NP * NP);
  float* P2  = alloc((size_t)NP * NP);
  float* BET = alloc((size_t)NP * MP);
  float* ALP = alloc((size_t)NP * MP);
  float* F1  = alloc((size_t)NP * XGW);
  float* F2  = alloc((size_t)NP * XGW);
  float* GC1 = alloc((size_t)NP * MP);
  float* GC2 = alloc((size_t)NP * MP);
  float* HC1 = alloc((size_t)NP * MP);
  float* HC2 = alloc((size_t)NP * MP);
  float* VH1 = alloc((size_t)NP * MP);
  float* VC1 = alloc((size_t)NP * MP);
  float* VH2 = alloc((size_t)NP * MP);
  float* VC2 = alloc((size_t)NP * MP);
  const size_t nFloats = off;
  int* ip = (int*)(w + off);
  int* CNT1 = ip;           // 16
  int* NOD1 = ip + 16;      // 16*128
  int* CNT2 = ip + 16 + 16 * 128;
  int* NOD2 = ip + 32 + 16 * 128;
  (void)ws_size; (void)n_in; (void)out_size;

  // ---- 0) zero all float staging (pads must be zero for padded WMMA tiles) ----
  zero_kernel<<<4096, 256, 0, stream>>>(w, (long)nFloats);

  // ---- 1) stage weights into WMMA-friendly padded + pair-swizzled layouts ----
  stage_wx_kernel<<<(4 * MU * EMB + 255) / 256, 256, 0, stream>>>(Wx_i, BXI);
  stage_wx_kernel<<<(4 * MU * EMB + 255) / 256, 256, 0, stream>>>(Wx_c, BXC);
  stage_wh_kernel<<<(5 * MU * EMB + 255) / 256, 256, 0, stream>>>(Wl_i, Wr_i, BHI);
  stage_wh_kernel<<<(5 * MU * EMB + 255) / 256, 256, 0, stream>>>(Wl_c, Wr_c, BHC);
  stage_wc_kernel<<<(MU * 4 * MU + 255) / 256, 256, 0, stream>>>(Wc, BWC);

  // ---- 2) level schedule of both trees ----
  levels_kernel<<<1, 32, 0, stream>>>(left1, right1, NND, CNT1, NOD1);
  levels_kernel<<<1, 32, 0, stream>>>(left2, right2, NND, CNT2, NOD2);

  // ---- 3) embedding gather ----
  embed_kernel<<<NP, MP, 0, stream>>>(sent1, left1, emb, pad_tok, X1);
  embed_kernel<<<NP, MP, 0, stream>>>(sent2, left2, emb, pad_tok, X2);

  // ---- 4) x-projections: XG = X @ BXI  [256x320]x[320x1280] ----
  gemm_f32_wmma<<<dim3(XGW / 64, NP / 128), 128, 0, stream>>>(X1, BXI, XG1, MP, MP, XGW, XGW);
  gemm_f32_wmma<<<dim3(XGW / 64, NP / 128), 128, 0, stream>>>(X2, BXI, XG2, MP, MP, XGW, XGW);

  // ---- level-parallel tree-LSTM scan ----
  auto run_scan = [&](const float* xg, const float* bx, const float* Bh,
                      const int* left, const int* right, const int* cnt,
                      const int* nod, float* C, float* H) {
    for (int lvl = 0; lvl < NLVL; ++lvl) {
      build_a_kernel<<<128, KH, 0, stream>>>(cnt, nod, lvl, left, right, H, ALV);
      gemm_f32_wmma<<<dim3(GW / 64, 1), 128, 0, stream>>>(ALV, Bh, GLV, KH, KH, GW, GW);
      gate_kernel<<<128, MP, 0, stream>>>(cnt, nod, lvl, left, right, xg, bx, GLV, C, H);
    }
  };

  // ---- 5) encode both sentences ----
  run_scan(XG1, bx_i, BHI, left1, right1, CNT1, NOD1, C1, H1);
  run_scan(XG2, bx_i, BHI, left2, right2, CNT2, NOD2, C2, H2);

  // ---- 6) attention: raw = H1 @ H2^T, softmax both ways, re-weight ----
  transpose_kernel<<<(MP * NP + 255) / 256, 256, 0, stream>>>(H2, H2T);
  hswz_kernel<<<(NP * MP + 255) / 256, 256, 0, stream>>>(H1, H1S);
  hswz_kernel<<<(NP * MP + 255) / 256, 256, 0, stream>>>(H2, H2S);
  gemm_f32_wmma<<<dim3(NP / 64, NP / 128), 128, 0, stream>>>(H1, H2T, RAW, MP, MP, NP, NP);
  softmax_rows_kernel<<<NP, 256, 0, stream>>>(RAW, P1);
  softmax_cols_kernel<<<NP, 256, 0, stream>>>(RAW, P2);
  gemm_f32_wmma<<<dim3(MP / 64, NP / 128), 128, 0, stream>>>(P1, H2S, BET, NP, NP, MP, MP);
  gemm_f32_wmma<<<dim3(MP / 64, NP / 128), 128, 0, stream>>>(P2, H1S, ALP, NP, NP, MP, MP);

  // ---- 7) compare features -> dense relu -> x-proj ----
  feat_kernel<<<NP, 256, 0, stream>>>(H1, BET, F1);
  feat_kernel<<<NP, 256, 0, stream>>>(H2, ALP, F2);
  gemm_f32_wmma<<<dim3(MP / 64, NP / 128), 128, 0, stream>>>(F1, BWC, GC1, XGW, XGW, MP, MP);
  gemm_f32_wmma<<<dim3(MP / 64, NP / 128), 128, 0, stream>>>(F2, BWC, GC2, XGW, XGW, MP, MP);
  relubias_kernel<<<NP, MP, 0, stream>>>(GC1, bc, HC1);
  relubias_kernel<<<NP, MP, 0, stream>>>(GC2, bc, HC2);
  gemm_f32_wmma<<<dim3(XGW / 64, NP / 128), 128, 0, stream>>>(HC1, BXC, XG1, MP, MP, XGW, XGW);
  gemm_f32_wmma<<<dim3(XGW / 64, NP / 128), 128, 0, stream>>>(HC2, BXC, XG2, MP, MP, XGW, XGW);

  // ---- 8) compare-composer scans ----
  run_scan(XG1, bx_c, BHC, left1, right1, CNT1, NOD1, VC1, VH1);
  run_scan(XG2, bx_c, BHC, left2, right2, CNT2, NOD2, VC2, VH2);

  // ---- 9) pooling + MLP head ----
  head_kernel<<<1, 512, 0, stream>>>(VH1, VH2, Wa1, ba1, Wa2, ba2, out);
}